// BlocksparseConv_46583215292898
// MI455X (gfx1250) — compile-verified
//
#include <hip/hip_runtime.h>

// CDNA5 / gfx1250 block-sparse 3x3 conv via implicit GEMM on V_WMMA_F32_16X16X4_F32.
// Memory-bound (~46 us floor @ 23.3 TB/s for ~1.07 GB fp32 traffic) -> keep exact
// fp32 math on the fp32 WMMA path. K-reduction is permuted to (kh,kw)-major /
// ic-minor so every inner-loop LDS address is (per-lane base) + (compile-time
// immediate), and each A/B fragment pair is one ds_load_b64.

#define IC_ 16
#define OC_ 16
#define H_ 512
#define W_ 512
#define KTOT 144           // 16 ic * 9 taps
#define NS   36            // KTOT / 4 WMMA steps
#define HT 4               // output rows per block
#define WT 64              // output cols per block
#define XR (HT + 2)        // 6 input rows (halo)
#define XC (WT + 2)        // 66 input cols (halo)
#define ICP 18             // ic-minor pad: stride 18 words -> conflict-free b64
#define KPAD 146           // K pad for weight LDS (146 mod 64 == 18)

typedef __attribute__((ext_vector_type(2))) float v2f;
typedef __attribute__((ext_vector_type(8))) float v8f;

__global__ void __launch_bounds__(512)
bsconv_wmma_f32_kernel(const float* __restrict__ x,
                       const float* __restrict__ wgt,
                       const float* __restrict__ bias,
                       const float* __restrict__ mask,
                       float* __restrict__ out)
{
    // xs[row][col][ic] ic-minor, padded; 6*66*18 floats = 27.8 KB
    __shared__ __align__(16) float xs[XR * XC * ICP];
    // wk2[oc][kperm], kperm = (kh*3+kw)*16 + ic, padded; 16*146 floats = 9.1 KB
    __shared__ __align__(16) float wk2[OC_ * KPAD];

    const int tid = threadIdx.x;
    const int w0  = blockIdx.x * WT;
    const int h0  = blockIdx.y * HT;
    const int n   = blockIdx.z;

    // ---- Stage masked weights, permuted to kperm = (kh*3+kw)*16 + ic, oc-major.
    for (int i = tid; i < OC_ * KTOT; i += 512) {
        int oc  = i / KTOT;             // wgt layout [OC][IC][KH][KW]
        int rem = i - oc * KTOT;        // rem = ic*9 + kh*3 + kw
        int ic  = rem / 9;
        int tap = rem - ic * 9;         // kh*3 + kw
        wk2[oc * KPAD + tap * 16 + ic] = wgt[i] * mask[i];
    }

    // ---- Stage x tile (halo, zero-padded), ic-minor in LDS, coalesced in w.
    for (int i = tid; i < IC_ * XR * XC; i += 512) {
        int ic = i / (XR * XC);
        int r  = (i / XC) % XR;
        int c  = i % XC;
        int gh = h0 + r - 1;
        int gw = w0 + c - 1;
        float v = 0.0f;
        if (gh >= 0 && gh < H_ && gw >= 0 && gw < W_) {
            v = x[((size_t)n * IC_ + ic) * ((size_t)H_ * W_) + (size_t)gh * W_ + gw];
        }
        xs[(r * XC + c) * ICP + ic] = v;
    }
    __syncthreads();

    // ---- Per-wave 16(OC) x 16(pixel) output tile.
    const int lane  = tid & 31;
    const int wv    = tid >> 5;      // 0..15
    const int ty    = wv >> 2;       // output row within tile: 0..3
    const int tx    = wv & 3;        // 16-wide column chunk: 0..3
    const int npix  = lane & 15;     // pixel col (B/N) == oc row (A/M)
    const int khalf = lane >> 4;     // 0: holds K{0,1}; 1: holds K{2,3}

    // Per-lane base pointers; all per-step deltas are compile-time immediates.
    const float* pa = &wk2[npix * KPAD + 2 * khalf];
    const float* pb = &xs[(ty * XC + (tx * 16 + npix)) * ICP + 2 * khalf];

    // Accumulator seeded with bias. C/D layout: VGPR j -> oc = j + 8*khalf.
    v8f acc;
#pragma unroll
    for (int j = 0; j < 8; ++j) acc[j] = bias[j + 8 * khalf];

#pragma unroll
    for (int s = 0; s < NS; ++s) {
        const int tap = s >> 2;              // kh*3 + kw  (fixed per slice)
        const int kh  = tap / 3;
        const int kw  = tap - kh * 3;
        const int icb = (s & 3) * 4;         // ic base of this slice

        v2f a = *(const v2f*)(pa + 4 * s);                       // K = 4s+2*khalf, +1
        v2f b = *(const v2f*)(pb + (kh * XC + kw) * ICP + icb);  // same K pair of patch

        acc = __builtin_amdgcn_wmma_f32_16x16x4_f32(
            /*neg_a=*/false, a, /*neg_b=*/false, b,
            /*c_mod=*/(short)0, acc, /*reuse_a=*/false, /*reuse_b=*/false);
    }

    // ---- Store: acc[j] -> out[n][oc = j + 8*khalf][h0+ty][w0 + tx*16 + npix]
    const size_t obase = ((size_t)n * OC_) * ((size_t)H_ * W_)
                       + (size_t)(h0 + ty) * W_ + (size_t)(w0 + tx * 16 + npix);
#pragma unroll
    for (int j = 0; j < 8; ++j) {
        const int oc = j + 8 * khalf;
        out[obase + (size_t)oc * ((size_t)H_ * W_)] = acc[j];
    }
}

extern "C" void kernel_launch(void* const* d_in, const int* in_sizes, int n_in,
                              void* d_out, int out_size, void* d_ws, size_t ws_size,
                              hipStream_t stream) {
    const float* x    = (const float*)d_in[0];   // [32,16,512,512]
    const float* wgt  = (const float*)d_in[1];   // [16,16,3,3]
    const float* bias = (const float*)d_in[2];   // [16]
    const float* mask = (const float*)d_in[3];   // [16,16,3,3]
    float* out = (float*)d_out;                  // [32,16,512,512]

    dim3 grid(W_ / WT, H_ / HT, 32);             // (8, 128, 32)
    dim3 block(512);                             // 16 wave32s
    bsconv_wmma_f32_kernel<<<grid, block, 0, stream>>>(x, wgt, bias, mask, out);
}